// HeteroGraphSAGE_65541200937531
// MI455X (gfx1250) — compile-verified
//
#include <hip/hip_runtime.h>

#define N_SI_C  100000
#define N_M_C   100000
#define N_INT_C 500000
#define E_C     500000
#define D_SI_C  128
#define D_INT_C 64
#define H1_C    32
#define H2_C    16

typedef float v2f __attribute__((ext_vector_type(2)));
typedef float v8f __attribute__((ext_vector_type(8)));

__device__ __forceinline__ v8f wmma4(v2f a, v2f b, v8f c) {
  // V_WMMA_F32_16X16X4_F32 : D = A(16x4) * B(4x16) + C(16x16), full f32
  return __builtin_amdgcn_wmma_f32_16x16x4_f32(false, a, false, b, (short)0, c,
                                               false, false);
}

// ---------------------------------------------------------------------------
// Prep: fold weights once per call (all tiny).
//   Wc_si = W1l_si + W1r_si ; Wc_m = W1l_m + W1r_m         [128x32]
//   Wr_comb = 0.5*(W1r_si2i + W1r_m2i)                     [64x32]
//   b1_comb = 0.5*(b1_si2i + b1_m2i)
//   v_si = W2l_si2i @ Wlin ; v_m = W2l_m2i @ Wlin          [32]
//   v_int = 0.5*(W2r_si2i + W2r_m2i) @ Wlin                [32]
//   c0 = 0.5*(b2_si2i + b2_m2i) @ Wlin + blin
// ---------------------------------------------------------------------------
__global__ void prep_kernel(
    const float* __restrict__ W1l_si, const float* __restrict__ W1r_si,
    const float* __restrict__ b1_si,
    const float* __restrict__ W1l_m, const float* __restrict__ W1r_m,
    const float* __restrict__ b1_m,
    const float* __restrict__ W1r_si2i, const float* __restrict__ W1r_m2i,
    const float* __restrict__ b1_si2i, const float* __restrict__ b1_m2i,
    const float* __restrict__ W2l_si2i, const float* __restrict__ W2l_m2i,
    const float* __restrict__ W2r_si2i, const float* __restrict__ W2r_m2i,
    const float* __restrict__ b2_si2i, const float* __restrict__ b2_m2i,
    const float* __restrict__ Wlin, const float* __restrict__ blin,
    float* __restrict__ Wc_si, float* __restrict__ Wc_m,
    float* __restrict__ Wr_comb,
    float* __restrict__ bc_si, float* __restrict__ bc_m,
    float* __restrict__ b1_comb,
    float* __restrict__ v_si, float* __restrict__ v_m,
    float* __restrict__ v_int, float* __restrict__ c0) {
  int t = threadIdx.x;
  for (int i = t; i < D_SI_C * H1_C; i += blockDim.x) {
    Wc_si[i] = W1l_si[i] + W1r_si[i];
    Wc_m[i]  = W1l_m[i]  + W1r_m[i];
  }
  for (int i = t; i < D_INT_C * H1_C; i += blockDim.x)
    Wr_comb[i] = 0.5f * (W1r_si2i[i] + W1r_m2i[i]);
  if (t < H1_C) {
    bc_si[t] = b1_si[t];
    bc_m[t] = b1_m[t];
    b1_comb[t] = 0.5f * (b1_si2i[t] + b1_m2i[t]);
    float s1 = 0.f, s2 = 0.f, s3 = 0.f;
    for (int j = 0; j < H2_C; j++) {
      float w = Wlin[j];
      s1 += W2l_si2i[t * H2_C + j] * w;
      s2 += W2l_m2i[t * H2_C + j] * w;
      s3 += 0.5f * (W2r_si2i[t * H2_C + j] + W2r_m2i[t * H2_C + j]) * w;
    }
    v_si[t] = s1; v_m[t] = s2; v_int[t] = s3;
  }
  if (t == 0) {
    float s = 0.f;
    for (int j = 0; j < H2_C; j++)
      s += 0.5f * (b2_si2i[j] + b2_m2i[j]) * Wlin[j];
    c0[0] = s + blin[0];
  }
}

// ---------------------------------------------------------------------------
// Per-dst in-degree counts (float, matches reference segment_sum of ones).
// ---------------------------------------------------------------------------
__global__ void count_kernel(const int* __restrict__ dst_si,
                             const int* __restrict__ dst_m,
                             float* __restrict__ cnt_si,
                             float* __restrict__ cnt_m) {
  int t = blockIdx.x * blockDim.x + threadIdx.x;
  if (t < E_C) {
    atomicAdd(&cnt_si[dst_si[t]], 1.0f);
  } else {
    t -= E_C;
    if (t < E_C) atomicAdd(&cnt_m[dst_m[t]], 1.0f);
  }
}

// ---------------------------------------------------------------------------
// Dual-output WMMA GEMM, K=128, N=32:
//   C1 = A @ B1                (projection for scatter)
//   C2 = relu(A @ B2 + bias2)  (fused self-loop SAGE layer)
// Block: 256 threads = 8 waves; wave -> 16 rows x 32 cols; block -> 128 rows.
// B staged transposed in LDS (stride BK=130: even + conflict-free).
// ---------------------------------------------------------------------------
__global__ __launch_bounds__(256) void gemm_dual128(
    const float* __restrict__ A, const float* __restrict__ B1,
    const float* __restrict__ B2, const float* __restrict__ bias2,
    float* __restrict__ C1, float* __restrict__ C2, int M) {
  const int K = 128, BK = 130;
  __shared__ float Bs1[32 * BK];
  __shared__ float Bs2[32 * BK];
  __shared__ float bsh[32];
  int tid = threadIdx.x;
  for (int i = tid; i < K * 32; i += 256) {
    int kk = i >> 5, col = i & 31;
    Bs1[col * BK + kk] = B1[i];
    Bs2[col * BK + kk] = B2[i];
  }
  if (tid < 32) bsh[tid] = bias2[tid];
  __syncthreads();

  int wave = tid >> 5, lane = tid & 31;
  int row0 = blockIdx.x * 128 + wave * 16;
  int half = lane >> 4;         // selects K pair {0,1} vs {2,3}
  int lrow = lane & 15;
  int arow = row0 + lrow;
  if (arow > M - 1) arow = M - 1;
  const float* Ap  = A + (size_t)arow * K + half * 2;
  const float* p1a = Bs1 + lrow * BK + half * 2;          // C1 cols 0..15
  const float* p1b = Bs1 + (lrow + 16) * BK + half * 2;   // C1 cols 16..31
  const float* p2a = Bs2 + lrow * BK + half * 2;
  const float* p2b = Bs2 + (lrow + 16) * BK + half * 2;

  v8f c1a = {}, c1b = {}, c2a = {}, c2b = {};
#pragma unroll 8
  for (int k = 0; k < K; k += 4) {
    v2f a   = *(const v2f*)(Ap + k);
    v2f b1a = *(const v2f*)(p1a + k);
    v2f b1b = *(const v2f*)(p1b + k);
    v2f b2a = *(const v2f*)(p2a + k);
    v2f b2b = *(const v2f*)(p2b + k);
    c1a = wmma4(a, b1a, c1a);
    c1b = wmma4(a, b1b, c1b);
    c2a = wmma4(a, b2a, c2a);
    c2b = wmma4(a, b2b, c2b);
  }

  int mbase = row0 + half * 8;
  int col = lrow;
#pragma unroll
  for (int r = 0; r < 8; r++) {
    int m = mbase + r;
    if (m < M) {
      size_t o = (size_t)m * 32 + col;
      C1[o]      = c1a[r];
      C1[o + 16] = c1b[r];
      C2[o]      = fmaxf(c2a[r] + bsh[col], 0.0f);
      C2[o + 16] = fmaxf(c2b[r] + bsh[col + 16], 0.0f);
    }
  }
}

// ---------------------------------------------------------------------------
// Single-output WMMA GEMM, K=64, N=32:  C = A @ B + bias  (no activation)
// ---------------------------------------------------------------------------
__global__ __launch_bounds__(256) void gemm_bias64(
    const float* __restrict__ A, const float* __restrict__ B,
    const float* __restrict__ bias,
    float* __restrict__ C, int M) {
  const int K = 64, BK = 66;
  __shared__ float Bs[32 * BK];
  __shared__ float bsh[32];
  int tid = threadIdx.x;
  for (int i = tid; i < K * 32; i += 256) {
    int kk = i >> 5, col = i & 31;
    Bs[col * BK + kk] = B[i];
  }
  if (tid < 32) bsh[tid] = bias[tid];
  __syncthreads();

  int wave = tid >> 5, lane = tid & 31;
  int row0 = blockIdx.x * 128 + wave * 16;
  int half = lane >> 4;
  int lrow = lane & 15;
  int arow = row0 + lrow;
  if (arow > M - 1) arow = M - 1;
  const float* Ap = A + (size_t)arow * K + half * 2;
  const float* pa = Bs + lrow * BK + half * 2;
  const float* pb = Bs + (lrow + 16) * BK + half * 2;

  v8f ca = {}, cb = {};
#pragma unroll
  for (int k = 0; k < K; k += 4) {
    v2f a  = *(const v2f*)(Ap + k);
    v2f ba = *(const v2f*)(pa + k);
    v2f bb = *(const v2f*)(pb + k);
    ca = wmma4(a, ba, ca);
    cb = wmma4(a, bb, cb);
  }

  int mbase = row0 + half * 8;
  int col = lrow;
#pragma unroll
  for (int r = 0; r < 8; r++) {
    int m = mbase + r;
    if (m < M) {
      size_t o = (size_t)m * 32 + col;
      C[o]      = ca[r] + bsh[col];
      C[o + 16] = cb[r] + bsh[col + 16];
    }
  }
}

// ---------------------------------------------------------------------------
// Layer-1 edge scatter: acc[dst] += 0.5 * p[src] / max(cnt[dst],1)
// One thread per (edge, feature); both relations in one launch.
// ---------------------------------------------------------------------------
__global__ void scatter1_kernel(
    const int* __restrict__ src_si, const int* __restrict__ dst_si,
    const int* __restrict__ src_m, const int* __restrict__ dst_m,
    const float* __restrict__ p_si, const float* __restrict__ p_m,
    const float* __restrict__ cnt_si, const float* __restrict__ cnt_m,
    float* __restrict__ acc) {
  int t = blockIdx.x * blockDim.x + threadIdx.x;
  int feat = t & 31;
  int e = t >> 5;
  const int* sp; const int* dp; const float* p; const float* cnt;
  if (e < E_C) {
    sp = src_si; dp = dst_si; p = p_si; cnt = cnt_si;
  } else {
    e -= E_C;
    if (e >= E_C) return;
    sp = src_m; dp = dst_m; p = p_m; cnt = cnt_m;
  }
  int s = sp[e], d = dp[e];
  float val = p[(size_t)s * 32 + feat] * (0.5f / fmaxf(cnt[d], 1.0f));
  atomicAdd(&acc[(size_t)d * 32 + feat], val);
}

// ---------------------------------------------------------------------------
// u[n] = h[n,:] . v   (layer-2 source features collapsed to scalars)
// ---------------------------------------------------------------------------
__global__ void udot_kernel(const float* __restrict__ h_si,
                            const float* __restrict__ h_m,
                            const float* __restrict__ v_si,
                            const float* __restrict__ v_m,
                            float* __restrict__ u_si,
                            float* __restrict__ u_m) {
  int i = blockIdx.x * blockDim.x + threadIdx.x;
  const float* h; const float* v; float* u;
  if (i < N_SI_C) {
    h = h_si + (size_t)i * 32; v = v_si; u = u_si + i;
  } else {
    i -= N_SI_C;
    if (i >= N_M_C) return;
    h = h_m + (size_t)i * 32; v = v_m; u = u_m + i;
  }
  float s = 0.f;
#pragma unroll
  for (int q = 0; q < 8; q++) {
    float4 a = ((const float4*)h)[q];
    float4 w = ((const float4*)v)[q];
    s += a.x * w.x + a.y * w.y + a.z * w.z + a.w * w.w;
  }
  *u = s;
}

// ---------------------------------------------------------------------------
// d_out[i] = relu(acc[i,:]) . v_int + c0     (h_int never materialized)
// ---------------------------------------------------------------------------
__global__ void finalize_kernel(const float* __restrict__ acc,
                                const float* __restrict__ v_int,
                                const float* __restrict__ c0,
                                float* __restrict__ out) {
  int i = blockIdx.x * blockDim.x + threadIdx.x;
  if (i >= N_INT_C) return;
  const float4* row = (const float4*)(acc + (size_t)i * 32);
  float s = 0.f;
#pragma unroll
  for (int q = 0; q < 8; q++) {
    float4 a = row[q];
    float4 w = ((const float4*)v_int)[q];
    s += fmaxf(a.x, 0.f) * w.x + fmaxf(a.y, 0.f) * w.y +
         fmaxf(a.z, 0.f) * w.z + fmaxf(a.w, 0.f) * w.w;
  }
  out[i] = s + c0[0];
}

// ---------------------------------------------------------------------------
// Layer-2 edge scatter: d_out[dst] += 0.5 * u[src] / max(cnt[dst],1)
// ---------------------------------------------------------------------------
__global__ void scatter2_kernel(
    const int* __restrict__ src_si, const int* __restrict__ dst_si,
    const int* __restrict__ src_m, const int* __restrict__ dst_m,
    const float* __restrict__ u_si, const float* __restrict__ u_m,
    const float* __restrict__ cnt_si, const float* __restrict__ cnt_m,
    float* __restrict__ out) {
  int t = blockIdx.x * blockDim.x + threadIdx.x;
  if (t < E_C) {
    int s = src_si[t], d = dst_si[t];
    atomicAdd(&out[d], u_si[s] * (0.5f / fmaxf(cnt_si[d], 1.0f)));
  } else {
    t -= E_C;
    if (t < E_C) {
      int s = src_m[t], d = dst_m[t];
      atomicAdd(&out[d], u_m[s] * (0.5f / fmaxf(cnt_m[d], 1.0f)));
    }
  }
}

extern "C" void kernel_launch(void* const* d_in, const int* in_sizes, int n_in,
                              void* d_out, int out_size, void* d_ws,
                              size_t ws_size, hipStream_t stream) {
  const float* x_si  = (const float*)d_in[0];
  const float* x_m   = (const float*)d_in[1];
  const float* x_int = (const float*)d_in[2];
  const float* W1l_si2i = (const float*)d_in[3];
  const float* W1r_si2i = (const float*)d_in[4];
  const float* b1_si2i  = (const float*)d_in[5];
  const float* W1l_m2i  = (const float*)d_in[6];
  const float* W1r_m2i  = (const float*)d_in[7];
  const float* b1_m2i   = (const float*)d_in[8];
  const float* W1l_si = (const float*)d_in[9];
  const float* W1r_si = (const float*)d_in[10];
  const float* b1_si  = (const float*)d_in[11];
  const float* W1l_m  = (const float*)d_in[12];
  const float* W1r_m  = (const float*)d_in[13];
  const float* b1_m   = (const float*)d_in[14];
  const float* W2l_si2i = (const float*)d_in[15];
  const float* W2r_si2i = (const float*)d_in[16];
  const float* b2_si2i  = (const float*)d_in[17];
  const float* W2l_m2i  = (const float*)d_in[18];
  const float* W2r_m2i  = (const float*)d_in[19];
  const float* b2_m2i   = (const float*)d_in[20];
  // d_in[21..26]: W2l_si/W2r_si/b2_si, W2l_m/W2r_m/b2_m — dead in reference.
  const float* Wlin = (const float*)d_in[27];
  const float* blin = (const float*)d_in[28];
  const int* src_si2i = (const int*)d_in[29];
  const int* dst_si2i = (const int*)d_in[30];
  const int* src_m2i  = (const int*)d_in[31];
  const int* dst_m2i  = (const int*)d_in[32];

  float* ws = (float*)d_ws;
  size_t o = 0;
  auto take = [&](size_t n) {
    size_t r = o;
    o += (n + 63) & ~(size_t)63;
    return r;
  };
  size_t o_wc_si   = take(D_SI_C * H1_C);
  size_t o_wc_m    = take(D_SI_C * H1_C);
  size_t o_wr_comb = take(D_INT_C * H1_C);
  size_t o_bc_si   = take(H1_C);
  size_t o_bc_m    = take(H1_C);
  size_t o_b1_comb = take(H1_C);
  size_t o_v_si    = take(H1_C);
  size_t o_v_m     = take(H1_C);
  size_t o_v_int   = take(H1_C);
  size_t o_c0      = take(1);
  size_t o_cnt_si  = take(N_INT_C);
  size_t o_cnt_m   = take(N_INT_C);
  size_t o_acc     = take((size_t)N_INT_C * H1_C);
  size_t o_p_si    = take((size_t)N_SI_C * H1_C);
  size_t o_p_m     = take((size_t)N_M_C * H1_C);
  size_t o_h_si    = take((size_t)N_SI_C * H1_C);
  size_t o_h_m     = take((size_t)N_M_C * H1_C);
  size_t o_u_si    = take(N_SI_C);
  size_t o_u_m     = take(N_M_C);
  (void)ws_size;

  // 1) fold weights
  prep_kernel<<<1, 256, 0, stream>>>(
      W1l_si, W1r_si, b1_si, W1l_m, W1r_m, b1_m,
      W1r_si2i, W1r_m2i, b1_si2i, b1_m2i,
      W2l_si2i, W2l_m2i, W2r_si2i, W2r_m2i, b2_si2i, b2_m2i, Wlin, blin,
      ws + o_wc_si, ws + o_wc_m, ws + o_wr_comb,
      ws + o_bc_si, ws + o_bc_m, ws + o_b1_comb,
      ws + o_v_si, ws + o_v_m, ws + o_v_int, ws + o_c0);

  // 2) zero 2*N_INT_C floats from cnt_si (pad gap: last 32 of cnt_m stay unzeroed)
  hipMemsetAsync(ws + o_cnt_si, 0, 2 * (size_t)N_INT_C * sizeof(float), stream);

  // 3) in-degree counts (both relations)
  count_kernel<<<(2 * E_C + 255) / 256, 256, 0, stream>>>(
      dst_si2i, dst_m2i, ws + o_cnt_si, ws + o_cnt_m);

  // 4) fused GEMMs for si / m source nodes: projection + self-loop SAGE
  int gsi = (N_SI_C + 127) / 128;
  gemm_dual128<<<gsi, 256, 0, stream>>>(
      x_si, W1l_si2i, ws + o_wc_si, ws + o_bc_si,
      ws + o_p_si, ws + o_h_si, N_SI_C);
  int gm = (N_M_C + 127) / 128;
  gemm_dual128<<<gm, 256, 0, stream>>>(
      x_m, W1l_m2i, ws + o_wc_m, ws + o_bc_m,
      ws + o_p_m, ws + o_h_m, N_M_C);

  // 5) x_int @ 0.5*(W1r_si2i+W1r_m2i) + b  -> acc (scatter accumulates on top)
  int gi = (N_INT_C + 127) / 128;
  gemm_bias64<<<gi, 256, 0, stream>>>(
      x_int, ws + o_wr_comb, ws + o_b1_comb, ws + o_acc, N_INT_C);

  // 6) layer-1 scatter-mean (pre-scaled by 0.5/cnt)
  long long w1 = 2LL * E_C * 32;
  scatter1_kernel<<<(int)((w1 + 255) / 256), 256, 0, stream>>>(
      src_si2i, dst_si2i, src_m2i, dst_m2i,
      ws + o_p_si, ws + o_p_m, ws + o_cnt_si, ws + o_cnt_m, ws + o_acc);

  // 7) collapse layer-2 source features to scalars: u = h . (W2l @ Wlin)
  udot_kernel<<<(N_SI_C + N_M_C + 255) / 256, 256, 0, stream>>>(
      ws + o_h_si, ws + o_h_m, ws + o_v_si, ws + o_v_m,
      ws + o_u_si, ws + o_u_m);

  // 8) d_out = relu(acc) . v_int + c0
  finalize_kernel<<<(N_INT_C + 255) / 256, 256, 0, stream>>>(
      ws + o_acc, ws + o_v_int, ws + o_c0, (float*)d_out);

  // 9) layer-2 scalar scatter-mean straight into d_out
  scatter2_kernel<<<(2 * E_C + 255) / 256, 256, 0, stream>>>(
      src_si2i, dst_si2i, src_m2i, dst_m2i,
      ws + o_u_si, ws + o_u_m, ws + o_cnt_si, ws + o_cnt_m, (float*)d_out);
}